// MORRMHA_24618752540970
// MI455X (gfx1250) — compile-verified
//
#include <hip/hip_runtime.h>
#include <hip/hip_bf16.h>

typedef float v2f __attribute__((ext_vector_type(2)));
typedef float v8f __attribute__((ext_vector_type(8)));

#define BS_   512
#define INF_  1024
#define OUTF_ 1024
#define P_    256
#define Q_    256

// Physics constants (double at compile time, truncated to float)
constexpr double kPI      = 3.14159265358979323846;
constexpr double kFWHM_d  = -4.0 * kPI * kPI * 2.0e-5 * 2.35 *
                            (1.0 / 1.55e-6 - 1.0 / (1.55e-6 - 1.0e-10));
constexpr float  kFWHM    = (float)kFWHM_d;
constexpr float  kA       = 0.8578f;
constexpr float  kR       = 0.8985f;
constexpr float  kTWO_AR  = 2.0f * kA * kR;                 // c = 2AR cos(phi)
constexpr float  kNUM     = kA * kA + kR * kR;              // A^2 + R^2
constexpr float  kDEN     = 1.0f + (kA * kR) * (kA * kR);   // 1 + (AR)^2
constexpr float  kINV2PI  = (float)(1.0 / (2.0 * kPI));     // v_cos operates on rev

// ---------------------------------------------------------------------------
// Setup kernel 1: s[q] table.  s[q] = ms[q]-mean (q<128), -(ms[q-128]-mean).
// ---------------------------------------------------------------------------
__global__ __launch_bounds__(256) void prep_s_kernel(
    const float* __restrict__ ms, float* __restrict__ sTab)
{
    __shared__ float red[256];
    const int t = threadIdx.x;
    red[t] = (t < 129) ? ms[t] : 0.0f;
    __syncthreads();
    #pragma unroll
    for (int off = 128; off > 0; off >>= 1) {
        if (t < off) red[t] += red[t + off];
        __syncthreads();
    }
    const float mean = red[0] * (1.0f / 129.0f);
    if (t < 128) {
        const float sm = ms[t] - mean;
        sTab[t]       = sm;
        sTab[t + 128] = -sm;
    }
}

// ---------------------------------------------------------------------------
// Setup kernel 2: biasTab[p,q] = FWHM * tanh(bias[p,q]) / (2*pi)
// (pre-scaled to "revolutions" so the hot loop feeds v_cos_f32 directly)
// ---------------------------------------------------------------------------
__global__ __launch_bounds__(256) void prep_bias_kernel(
    const float* __restrict__ bias, float* __restrict__ biasTab)
{
    const int idx = blockIdx.x * 256 + threadIdx.x;
    biasTab[idx] = (kFWHM * kINV2PI) * tanhf(bias[idx]);
}

// ---------------------------------------------------------------------------
// Main kernel: one wave owns a 16(b) x 16(p,i) tile, loops over q = 0..255.
// WMMA f32 16x16x4 computes phi/(2*pi) (the 1/2pi factor is folded into the
// A fragment), then cos -> rational transfer function elementwise, weighted
// by s[q] and accumulated.  EXEC all-ones throughout (no divergence).
// PRE = true : bias/scale tables precomputed in workspace (fast path)
// PRE = false: fully fused fallback (no workspace needed)
// ---------------------------------------------------------------------------
template <bool PRE>
__global__ __launch_bounds__(256) void morr_main_kernel(
    const float* __restrict__ x,       // (512, 1024)
    const float* __restrict__ weight,  // (256, 256, 4)
    const float* __restrict__ ms,      // (129,)
    const float* __restrict__ bias,    // (256, 256)
    const float* __restrict__ sTab,    // (256,)   [PRE]
    const float* __restrict__ biasTab, // (256,256)[PRE]
    float* __restrict__ out)           // (512, 1024)
{
    float mean = 0.0f;
    if (!PRE) {
        float sum = 0.0f;
        #pragma unroll 1
        for (int t = 0; t < 129; ++t) sum += ms[t];
        mean = sum * (1.0f / 129.0f);
    }

    // tile assignment: 32 b-tiles x 64 column-tiles = 2048 waves
    const int lane  = threadIdx.x & 31;
    const int wv    = threadIdx.x >> 5;
    const int W     = blockIdx.x * 8 + wv;
    const int b0    = (W & 31) << 4;          // b tile origin
    const int col0  = (W >> 5) << 4;          // output column origin = 4*p0
    const int p0    = col0 >> 2;

    const int half  = lane >> 4;              // 0: K={0,1}, 1: K={2,3}
    const int l     = lane & 15;
    const int j0    = half << 1;

    // B-fragment gather (per-lane constants): column n -> (p, i)
    const int n     = l;
    const int p     = p0 + (n >> 2);
    const int i     = n & 3;
    const int idx0  = (i - j0) & 3;           // K = j0
    const int idx1  = (i - j0 - 1) & 3;       // K = j0 + 1

    const float* xrow = x + (b0 + l) * INF_ + j0;   // A fragment: row b0+l
    const float* wrow = weight + p * (Q_ * 4);      // advances 4 floats per q
    const float* brow = PRE ? (biasTab + p * Q_) : (bias + p * Q_);

    v8f acc = {0.f, 0.f, 0.f, 0.f, 0.f, 0.f, 0.f, 0.f};

    #pragma unroll 2
    for (int q = 0; q < Q_; ++q) {
        // A: xq = x^2 / (2*pi) for (row = b0+l, K = j0, j0+1)
        v2f xa = *(const v2f*)(xrow + q * 4);
        v2f Af;
        Af.x = (xa.x * xa.x) * kINV2PI;
        Af.y = (xa.y * xa.y) * kINV2PI;

        // B: circulant |w| gather for column n, K = j0, j0+1
        const float* wq = wrow + q * 4;
        v2f Bf;
        Bf.x = __builtin_fabsf(wq[idx0]);
        Bf.y = __builtin_fabsf(wq[idx1]);

        // phi/(2pi) (16x16) = A(16x4) x B(4x16), C = 0
        v8f cz = {0.f, 0.f, 0.f, 0.f, 0.f, 0.f, 0.f, 0.f};
        v8f d = __builtin_amdgcn_wmma_f32_16x16x4_f32(
                    false, Af, false, Bf, (short)0, cz, false, false);

        // per-(p,q) bias (same for all 8 rows of this lane) and uniform s[q]
        float bq, sq;                         // bq is pre-scaled by 1/2pi
        if (PRE) {
            bq = brow[q];
            sq = sTab[q];
        } else {
            bq = (kFWHM * kINV2PI) * tanhf(brow[q]);
            const float msv = (q < 128) ? ms[q] : ms[q - 128];
            sq = msv - mean;
            if (q >= 128) sq = -sq;
        }

        #pragma unroll
        for (int r = 0; r < 8; ++r) {
            const float phin = d[r] - bq;                        // phi/(2*pi)
            const float c    = kTWO_AR * __builtin_amdgcn_cosf(phin); // cos(phi)
            const float tr   = (kNUM - c) * __builtin_amdgcn_rcpf(kDEN - c);
            acc[r] = fmaf(sq, tr, acc[r]);
        }
    }

    // store: VGPR r -> row b0 + r + 8*half, col col0 + l
    float* orow = out + (b0 + half * 8) * OUTF_ + col0 + l;
    #pragma unroll
    for (int r = 0; r < 8; ++r) {
        orow[r * OUTF_] = acc[r];
    }
}

extern "C" void kernel_launch(void* const* d_in, const int* in_sizes, int n_in,
                              void* d_out, int out_size, void* d_ws, size_t ws_size,
                              hipStream_t stream) {
    (void)in_sizes; (void)n_in; (void)out_size;
    const float* x      = (const float*)d_in[0];
    const float* weight = (const float*)d_in[1];
    const float* ms     = (const float*)d_in[2];
    const float* bias   = (const float*)d_in[3];
    float* out          = (float*)d_out;

    // workspace layout: sTab[256] floats, then biasTab[65536] floats
    const size_t need = (256 + (size_t)P_ * Q_) * sizeof(float);
    dim3 grid(256), block(256);

    if (ws_size >= need) {
        float* sTab    = (float*)d_ws;
        float* biasTab = sTab + 256;
        hipLaunchKernelGGL(prep_s_kernel, dim3(1), block, 0, stream, ms, sTab);
        hipLaunchKernelGGL(prep_bias_kernel, dim3(P_ * Q_ / 256), block, 0, stream,
                           bias, biasTab);
        hipLaunchKernelGGL((morr_main_kernel<true>), grid, block, 0, stream,
                           x, weight, ms, bias, sTab, biasTab, out);
    } else {
        hipLaunchKernelGGL((morr_main_kernel<false>), grid, block, 0, stream,
                           x, weight, ms, bias, nullptr, nullptr, out);
    }
}